// Stage1ConsistencyLoss_66331474919882
// MI455X (gfx1250) — compile-verified
//
#include <hip/hip_runtime.h>

#define NROWS 4096
#define DDIM  1024
#define BM    128                       // macro-tile is BM x BM
#define BK    32
#define NB    (NROWS / BM)              // 32 block-rows
#define NTILES (NB * (NB + 1) / 2)      // 528 upper-tri 128x128 tiles
#define MARGIN 0.3f
#define EPSN   1e-8f

typedef float v2f __attribute__((ext_vector_type(2)));
typedef float v8f __attribute__((ext_vector_type(8)));

// ---------------------------------------------------------------------------
// Kernel 1: rnorm[i] = 1 / max(||X[i,:]||, eps)
// ---------------------------------------------------------------------------
__global__ void __launch_bounds__(128)
rnorm_kernel(const float* __restrict__ X, float* __restrict__ rnorm) {
    __shared__ float red[128];
    const int row = blockIdx.x;
    const int t   = threadIdx.x;
    const float* p = X + (size_t)row * DDIM;
    float ss = 0.0f;
    #pragma unroll
    for (int e = 0; e < DDIM; e += 128) {
        float v = p[t + e];
        ss += v * v;
    }
    red[t] = ss;
    __syncthreads();
    for (int s = 64; s > 0; s >>= 1) {
        if (t < s) red[t] += red[t + s];
        __syncthreads();
    }
    if (t == 0) rnorm[row] = 1.0f / fmaxf(sqrtf(red[0]), EPSN);
}

// ---------------------------------------------------------------------------
// Kernel 2: upper-triangular blocked Gram matrix via V_WMMA_F32_16X16X4_F32.
// 128x128 macro-tile, 8 waves, each wave computes 32x64 (2x4 WMMA tiles).
// Tiles staged global->LDS with GLOBAL_LOAD_ASYNC_TO_LDS_B128 (no VGPR
// round-trip), tracked via ASYNCcnt. Fused margin-loss epilogue +
// deterministic per-block reduction.
// ---------------------------------------------------------------------------
__global__ void __launch_bounds__(256)
simloss_kernel(const float* __restrict__ X, const int* __restrict__ labels,
               const float* __restrict__ rnorm, float* __restrict__ partials) {
    __shared__ float As[BM][BK + 4];   // stride 36: conflict-free, 16B-aligned
    __shared__ float Bs[BM][BK + 4];

    const int t = threadIdx.x;

    // linear tile id -> (bi, bj) with bi <= bj
    int rem = blockIdx.x;
    int bi = 0;
    while (rem >= NB - bi) { rem -= NB - bi; ++bi; }
    const int bj = bi + rem;
    const int rowBase = bi * BM;
    const int colBase = bj * BM;

    const int lane  = t & 31;
    const int wid   = t >> 5;        // 0..7
    const int waveM = wid & 3;       // 4 waves down M: 32-row slab
    const int waveN = wid >> 2;      // 2 waves across N: 64-col slab
    const int hl    = lane >> 4;     // lane half: selects K pair / M offset
    const int l16   = lane & 15;

    v8f acc[2][4] = {};              // [mt][nt] 16x16 tiles

    const int aRow = waveM * 32 + l16;   // +mt*16
    const int bRow = waveN * 64 + l16;   // +nt*16

    // --- precompute async staging addresses -------------------------------
    // slot v = t + s*256  ->  row = (t>>3) + s*32,  c4 = (t&7)*4  (invariant)
    const int stRow = t >> 3;
    const int stC4  = (t & 7) << 2;
    unsigned ldsA[4], ldsB[4];
    const float* gA[4];
    const float* gB[4];
    #pragma unroll
    for (int s = 0; s < 4; ++s) {
        const int r = stRow + s * 32;
        ldsA[s] = (unsigned)(unsigned long long)(uintptr_t)&As[r][stC4];
        ldsB[s] = (unsigned)(unsigned long long)(uintptr_t)&Bs[r][stC4];
        gA[s] = X + (size_t)(rowBase + r) * DDIM + stC4;
        gB[s] = X + (size_t)(colBase + r) * DDIM + stC4;
    }

    for (int k0 = 0; k0 < DDIM; k0 += BK) {
        __syncthreads();   // previous iteration's LDS reads complete
        // Async DMA: 128x32 A tile and B tile, 16B per lane per op.
        #pragma unroll
        for (int s = 0; s < 4; ++s) {
            asm volatile("global_load_async_to_lds_b128 %0, %1, off"
                         :: "v"(ldsA[s]), "v"(gA[s] + k0) : "memory");
            asm volatile("global_load_async_to_lds_b128 %0, %1, off"
                         :: "v"(ldsB[s]), "v"(gB[s] + k0) : "memory");
        }
        asm volatile("s_wait_asynccnt 0x0" ::: "memory");
        __syncthreads();   // tile visible to all waves

        #pragma unroll
        for (int kk = 0; kk < BK; kk += 4) {
            const int ko = kk + hl * 2;
            v2f a[2], b[4];
            #pragma unroll
            for (int mt = 0; mt < 2; ++mt) {        // A 16x4: M=l16, K=hl*2+{0,1}
                a[mt].x = As[aRow + mt * 16][ko];
                a[mt].y = As[aRow + mt * 16][ko + 1];
            }
            #pragma unroll
            for (int nt = 0; nt < 4; ++nt) {        // B 4x16: N=l16, K=hl*2+{0,1}
                b[nt].x = Bs[bRow + nt * 16][ko];
                b[nt].y = Bs[bRow + nt * 16][ko + 1];
            }
            #pragma unroll
            for (int mt = 0; mt < 2; ++mt)
                #pragma unroll
                for (int nt = 0; nt < 4; ++nt)
                    acc[mt][nt] = __builtin_amdgcn_wmma_f32_16x16x4_f32(
                        false, a[mt], false, b[nt], (short)0, acc[mt][nt],
                        false, false);
        }
    }

    // Epilogue: sim = gram * rnorm_i * rnorm_j; margin loss; i<j mask.
    float rnj[4];
    int   lj[4];
    #pragma unroll
    for (int nt = 0; nt < 4; ++nt) {
        int j = colBase + waveN * 64 + nt * 16 + l16;   // C/D: N = lane&15
        rnj[nt] = rnorm[j];
        lj[nt]  = labels[j];
    }
    float sum = 0.0f;
    #pragma unroll
    for (int mt = 0; mt < 2; ++mt) {
        #pragma unroll
        for (int r = 0; r < 8; ++r) {
            const int i = rowBase + waveM * 32 + mt * 16 + r + hl * 8; // M=r+8*half
            const float rni = rnorm[i];
            const int   li  = labels[i];
            #pragma unroll
            for (int nt = 0; nt < 4; ++nt) {
                const int j = colBase + waveN * 64 + nt * 16 + l16;
                float sim  = acc[mt][nt][r] * rni * rnj[nt];
                float loss = (lj[nt] == li) ? (1.0f - sim)
                                            : fmaxf(sim - MARGIN, 0.0f);
                if (j > i) sum += loss;
            }
        }
    }

    // Deterministic block reduction (reuse As as scratch)
    __syncthreads();
    float* red = &As[0][0];
    red[t] = sum;
    __syncthreads();
    for (int s = 128; s > 0; s >>= 1) {
        if (t < s) red[t] += red[t + s];
        __syncthreads();
    }
    if (t == 0) partials[blockIdx.x] = red[0];
}

// ---------------------------------------------------------------------------
// Kernel 3: deterministic final reduction + mean
// ---------------------------------------------------------------------------
__global__ void __launch_bounds__(256)
finalize_kernel(const float* __restrict__ partials, float* __restrict__ out) {
    __shared__ float red[256];
    const int t = threadIdx.x;
    float s = 0.0f;
    for (int i = t; i < NTILES; i += 256) s += partials[i];
    red[t] = s;
    __syncthreads();
    for (int k = 128; k > 0; k >>= 1) {
        if (t < k) red[t] += red[t + k];
        __syncthreads();
    }
    if (t == 0) {
        const float count = (float)((long long)NROWS * (NROWS - 1) / 2); // 8386560
        out[0] = red[0] / count;
    }
}

extern "C" void kernel_launch(void* const* d_in, const int* in_sizes, int n_in,
                              void* d_out, int out_size, void* d_ws, size_t ws_size,
                              hipStream_t stream) {
    const float* X      = (const float*)d_in[0];
    const int*   labels = (const int*)d_in[1];
    float* rnorm    = (float*)d_ws;          // 4096 floats
    float* partials = rnorm + NROWS;         // 528 floats
    float* out      = (float*)d_out;

    rnorm_kernel<<<NROWS, 128, 0, stream>>>(X, rnorm);
    simloss_kernel<<<NTILES, 256, 0, stream>>>(X, labels, rnorm, partials);
    finalize_kernel<<<1, 256, 0, stream>>>(partials, out);
}